// densityPropGRUCell_18476949308043
// MI455X (gfx1250) — compile-verified
//
#include <hip/hip_runtime.h>
#include <hip/hip_bf16.h>

typedef __attribute__((ext_vector_type(2))) float v2f;
typedef __attribute__((ext_vector_type(8))) float v8f;

#define UDIM 256
#define BSZ  128
#define UU   65536
#define VEC_STRIDE 4096
#define SLOT_S  0
#define SLOT_R  256
#define SLOT_Z  512
#define SLOT_D  768
#define SLOT_GZ 1024
#define SLOT_GR 1280
#define SLOT_GH 1536
#define SLOT_DZ 1792
#define SLOT_DR 2048
#define SLOT_DH 2304
#define SLOT_SC 2560
#define SR_OFF  ((size_t)BSZ * VEC_STRIDE)   // floats

__device__ __forceinline__ float softplusf_(float x) { return log1pf(expf(x)); }
__device__ __forceinline__ float sigmoidf_(float x) { return 1.0f / (1.0f + expf(-x)); }

__device__ __forceinline__ v8f wmma4(v2f a, v2f b, v8f c) {
  // D = A(16x4 f32) * B(4x16 f32) + C(16x16 f32)
  return __builtin_amdgcn_wmma_f32_16x16x4_f32(
      /*neg_a=*/false, a, /*neg_b=*/false, b,
      /*c_mod=*/(short)0, c, /*reuse_a=*/false, /*reuse_b=*/false);
}

__device__ __forceinline__ float block_sum256(float v, float* red) {
  int t = threadIdx.x;
  red[t] = v;
  __syncthreads();
  for (int off = 128; off > 0; off >>= 1) {
    if (t < off) red[t] += red[t + off];
    __syncthreads();
  }
  float s = red[0];
  __syncthreads();
  return s;
}

// ---------------------------------------------------------------------------
// Kernel 1: gate means, activations, scalars, diag vectors, mu_out
// grid: (BSZ), block: 256
// ---------------------------------------------------------------------------
__global__ __launch_bounds__(256) void gates_kernel(
    const float* __restrict__ x_in, const float* __restrict__ s_in,
    const float* __restrict__ Sg,
    const float* __restrict__ Uz, const float* __restrict__ uzs,
    const float* __restrict__ Wz, const float* __restrict__ wzs,
    const float* __restrict__ Ur, const float* __restrict__ urs,
    const float* __restrict__ Wr, const float* __restrict__ wrs,
    const float* __restrict__ Uh, const float* __restrict__ uhs,
    const float* __restrict__ Wh, const float* __restrict__ whs,
    float* __restrict__ mu_out, float* __restrict__ ws)
{
  __shared__ float lx[UDIM], ls[UDIM], lsr[UDIM], red[256];
  const int b = blockIdx.x;
  const int t = threadIdx.x;

  lx[t] = x_in[(size_t)b * UDIM + t];
  ls[t] = s_in[(size_t)b * UDIM + t];
  __syncthreads();

  float mz = 0.f, mr = 0.f;
  for (int k = 0; k < UDIM; ++k) {
    float xk = lx[k], sk = ls[k];
    mz += xk * Uz[k * UDIM + t] + sk * Wz[k * UDIM + t];
    mr += xk * Ur[k * UDIM + t] + sk * Wr[k * UDIM + t];
  }
  float z = sigmoidf_(mz);
  float r = sigmoidf_(mr);
  lsr[t] = ls[t] * r;
  __syncthreads();

  float mh = 0.f;
  for (int k = 0; k < UDIM; ++k)
    mh += lx[k] * Uh[k * UDIM + t] + lsr[k] * Wh[k * UDIM + t];
  float h = tanhf(mh);

  float ax  = block_sum256(lx[t] * lx[t], red);
  float as2 = block_sum256(ls[t] * ls[t], red);
  float asr = block_sum256(lsr[t] * lsr[t], red);
  float trS = block_sum256(Sg[(size_t)b * UU + (size_t)t * UDIM + t], red);

  float* vb = ws + (size_t)b * VEC_STRIDE;
  vb[SLOT_S + t]  = ls[t];
  vb[SLOT_R + t]  = r;
  vb[SLOT_Z + t]  = z;
  vb[SLOT_D + t]  = ls[t] - h;
  vb[SLOT_GZ + t] = z * (1.f - z);
  vb[SLOT_GR + t] = r * (1.f - r);
  vb[SLOT_GH + t] = 1.f - h * h;
  vb[SLOT_DZ + t] = (as2 + trS) * softplusf_(wzs[t]) + ax * softplusf_(uzs[t]);
  vb[SLOT_DR + t] = (as2 + trS) * softplusf_(wrs[t]) + ax * softplusf_(urs[t]);
  if (t == 0) { vb[SLOT_SC + 0] = ax; vb[SLOT_SC + 1] = asr; }

  mu_out[(size_t)b * UDIM + t] = z * ls[t] + (1.f - z) * h;
}

// ---------------------------------------------------------------------------
// Kernel 2: trace(sigma_g) from diagonals + h-gate diag vector
// grid: (BSZ), block: 256
// ---------------------------------------------------------------------------
__global__ __launch_bounds__(256) void traceg_kernel(
    const float* __restrict__ Sg, const float* __restrict__ uhs,
    const float* __restrict__ whs, float* __restrict__ ws)
{
  __shared__ float red[256];
  const int b = blockIdx.x;
  const int t = threadIdx.x;
  float* vb = ws + (size_t)b * VEC_STRIDE;
  const float* SR = ws + SR_OFF + (size_t)b * UU;

  float Sd  = Sg[(size_t)b * UU + (size_t)t * UDIM + t];
  float SRd = SR[(size_t)t * UDIM + t];
  float si = vb[SLOT_S + t], ri = vb[SLOT_R + t];
  float gii = SRd * (Sd + si * si) + ri * ri * Sd;   // sigma_g diagonal
  float trG = block_sum256(gii, red);

  float ax = vb[SLOT_SC + 0], asr = vb[SLOT_SC + 1];
  vb[SLOT_DH + t] = (asr + trG) * softplusf_(whs[t]) + ax * softplusf_(uhs[t]);
}

// ---------------------------------------------------------------------------
// Kernel 3: fused  C = W^T * A * W  (+ diag, + outer(g,g), [+ final combine])
// A = Sigma_state (MODE 0/1) or on-the-fly sigma_g (MODE 2).
// grid: (4 col-blocks, BSZ), block: 256 (8 waves); each block: 256 rows x 64 cols
// Phase 1: M1 = A * W[:,block]  -> LDS (256 x 68 padded)
// Phase 2: C  = W^T * M1        -> epilogue -> global
// MODE is a template parameter so the hot k-loops have no runtime branches.
// ---------------------------------------------------------------------------
#define M1LD 68
template <int MODE>
__global__ __launch_bounds__(256) void fused_wtsw_kernel(
    const float* __restrict__ Sg, const float* __restrict__ W,
    float* __restrict__ ws, float* __restrict__ outSigma,
    int g_slot, int d_slot)
{
  __shared__ float lds_m1[UDIM * M1LD];
  __shared__ float lv_s[UDIM], lv_r[UDIM], lv_g[UDIM], lv_dv[UDIM],
                   lv_z[UDIM], lv_d[UDIM];

  const int b  = blockIdx.y;
  const int l0 = blockIdx.x * 64;
  const int t  = threadIdx.x;

  const float* vb = ws + (size_t)b * VEC_STRIDE;
  lv_s[t]  = vb[SLOT_S + t];
  lv_r[t]  = vb[SLOT_R + t];
  lv_g[t]  = vb[g_slot + t];
  lv_dv[t] = vb[d_slot + t];
  lv_z[t]  = vb[SLOT_Z + t];
  lv_d[t]  = vb[SLOT_D + t];
  __syncthreads();

  const float* Sb  = Sg + (size_t)b * UU;
  const float* SRb = ws + SR_OFF + (size_t)b * UU;

  const int wave = t >> 5, lane = t & 31, lane16 = lane & 15;
  const int hi = lane >> 4;       // 0: lanes 0-15, 1: lanes 16-31
  const int kOff = hi * 2;        // K offset within a 16x4 A / 4x16 B fragment
  const int m0 = wave * 32;       // this wave's 32-row strip

  const v8f vzero = {0.f, 0.f, 0.f, 0.f, 0.f, 0.f, 0.f, 0.f};
  v8f acc[2][4];
#pragma unroll
  for (int i = 0; i < 2; ++i)
#pragma unroll
    for (int j = 0; j < 4; ++j) acc[i][j] = vzero;

  // -------------------- Phase 1: M1 = A * W[:,block] --------------------
  const int rowA0 = m0 + lane16;
  const int rowA1 = m0 + 16 + lane16;
  const float srow0 = lv_s[rowA0], srow1 = lv_s[rowA1];
  const float rrow0 = lv_r[rowA0], rrow1 = lv_r[rowA1];

  for (int k0 = 0; k0 < UDIM; k0 += 4) {
    const int kk = k0 + kOff;
    v2f s0 = *(const v2f*)(Sb + (size_t)rowA0 * UDIM + kk);
    v2f s1 = *(const v2f*)(Sb + (size_t)rowA1 * UDIM + kk);
    v2f a0, a1;
    if (MODE == 2) {  // build sigma_g[j,k] = SR*(S + s_j s_k) + r_j r_k * S
      v2f q0 = *(const v2f*)(SRb + (size_t)rowA0 * UDIM + kk);
      v2f q1 = *(const v2f*)(SRb + (size_t)rowA1 * UDIM + kk);
      float sk0 = lv_s[kk], sk1 = lv_s[kk + 1];
      float rk0 = lv_r[kk], rk1 = lv_r[kk + 1];
      a0.x = q0.x * (s0.x + srow0 * sk0) + rrow0 * rk0 * s0.x;
      a0.y = q0.y * (s0.y + srow0 * sk1) + rrow0 * rk1 * s0.y;
      a1.x = q1.x * (s1.x + srow1 * sk0) + rrow1 * rk0 * s1.x;
      a1.y = q1.y * (s1.y + srow1 * sk1) + rrow1 * rk1 * s1.y;
    } else {
      a0 = s0; a1 = s1;
    }
    v2f bf[4];
#pragma unroll
    for (int c = 0; c < 4; ++c) {
      int col = l0 + c * 16 + lane16;
      bf[c].x = W[(size_t)kk * UDIM + col];
      bf[c].y = W[(size_t)(kk + 1) * UDIM + col];
    }
#pragma unroll
    for (int c = 0; c < 4; ++c) {
      acc[0][c] = wmma4(a0, bf[c], acc[0][c]);
      acc[1][c] = wmma4(a1, bf[c], acc[1][c]);
    }
  }

  // spill M1 strip to LDS (row j in 0..255, local col 0..63)
#pragma unroll
  for (int rt = 0; rt < 2; ++rt)
#pragma unroll
    for (int c = 0; c < 4; ++c)
#pragma unroll
      for (int v = 0; v < 8; ++v) {
        int row = m0 + rt * 16 + v + hi * 8;
        lds_m1[row * M1LD + c * 16 + lane16] = acc[rt][c][v];
      }
  __syncthreads();

  // -------------------- Phase 2: C = W^T * M1 --------------------
#pragma unroll
  for (int i = 0; i < 2; ++i)
#pragma unroll
    for (int j = 0; j < 4; ++j) acc[i][j] = vzero;

  const int i0 = m0;
  for (int k0 = 0; k0 < UDIM; k0 += 4) {
    const int kk = k0 + kOff;
    v2f a0, a1;
    a0.x = W[(size_t)kk * UDIM + i0 + lane16];
    a0.y = W[(size_t)(kk + 1) * UDIM + i0 + lane16];
    a1.x = W[(size_t)kk * UDIM + i0 + 16 + lane16];
    a1.y = W[(size_t)(kk + 1) * UDIM + i0 + 16 + lane16];
    v2f bf[4];
#pragma unroll
    for (int c = 0; c < 4; ++c) {
      bf[c].x = lds_m1[kk * M1LD + c * 16 + lane16];
      bf[c].y = lds_m1[(kk + 1) * M1LD + c * 16 + lane16];
    }
#pragma unroll
    for (int c = 0; c < 4; ++c) {
      acc[0][c] = wmma4(a0, bf[c], acc[0][c]);
      acc[1][c] = wmma4(a1, bf[c], acc[1][c]);
    }
  }

  // -------------------- Epilogue --------------------
  float* SRout = ws + SR_OFF;
#pragma unroll
  for (int rt = 0; rt < 2; ++rt)
#pragma unroll
    for (int c = 0; c < 4; ++c)
#pragma unroll
      for (int v = 0; v < 8; ++v) {
        int i = i0 + rt * 16 + v + hi * 8;
        int l = l0 + c * 16 + lane16;
        float core = acc[rt][c][v] + ((i == l) ? lv_dv[i] : 0.f);
        float val = core * lv_g[i] * lv_g[l];
        size_t pos = (size_t)b * UU + (size_t)i * UDIM + l;
        if (MODE == 0) {
          outSigma[pos] = val;                    // Sigma_z
        } else if (MODE == 1) {
          SRout[pos] = val;                       // Sigma_r
        } else {
          float H  = val;                         // Sigma_h
          float SZ = outSigma[pos];               // Sigma_z (written by MODE 0)
          float Sv = Sb[(size_t)i * UDIM + l];    // Sigma_state
          float zi = lv_z[i], zl = lv_z[l];
          float res = SZ * (Sv + H + lv_d[i] * lv_d[l])
                    + zi * zl * Sv
                    + (1.f - zi) * (1.f - zl) * H;
          outSigma[pos] = res;                    // Sigma_out, in place
        }
      }
}

// ---------------------------------------------------------------------------
extern "C" void kernel_launch(void* const* d_in, const int* in_sizes, int n_in,
                              void* d_out, int out_size, void* d_ws, size_t ws_size,
                              hipStream_t stream) {
  const float* x   = (const float*)d_in[0];
  const float* s   = (const float*)d_in[1];
  const float* Sg  = (const float*)d_in[2];
  const float* Uz  = (const float*)d_in[3];
  const float* uzs = (const float*)d_in[4];
  const float* Wz  = (const float*)d_in[5];
  const float* wzs = (const float*)d_in[6];
  const float* Ur  = (const float*)d_in[7];
  const float* urs = (const float*)d_in[8];
  const float* Wr  = (const float*)d_in[9];
  const float* wrs = (const float*)d_in[10];
  const float* Uh  = (const float*)d_in[11];
  const float* uhs = (const float*)d_in[12];
  const float* Wh  = (const float*)d_in[13];
  const float* whs = (const float*)d_in[14];

  float* out      = (float*)d_out;
  float* mu       = out;                       // [B,U]
  float* outSigma = out + (size_t)BSZ * UDIM;  // [B,U,U]
  float* ws       = (float*)d_ws;              // needs ~34 MB

  gates_kernel<<<BSZ, 256, 0, stream>>>(x, s, Sg, Uz, uzs, Wz, wzs,
                                        Ur, urs, Wr, wrs, Uh, uhs, Wh, whs,
                                        mu, ws);
  dim3 g(4, BSZ);
  fused_wtsw_kernel<0><<<g, 256, 0, stream>>>(Sg, Wz, ws, outSigma, SLOT_GZ, SLOT_DZ);
  fused_wtsw_kernel<1><<<g, 256, 0, stream>>>(Sg, Wr, ws, outSigma, SLOT_GR, SLOT_DR);
  traceg_kernel<<<BSZ, 256, 0, stream>>>(Sg, uhs, whs, ws);
  fused_wtsw_kernel<2><<<g, 256, 0, stream>>>(Sg, Wh, ws, outSigma, SLOT_GH, SLOT_DH);
}